// VectorQuantizer_20478404067972
// MI455X (gfx1250) — compile-verified
//
#include <hip/hip_runtime.h>

// ---------------------------------------------------------------------------
// VQ-VAE vector quantizer for MI455X (gfx1250).
//   B=64, D=64, H=W=32  -> N = 65536 rows of dim 64;  K = 1024 codes.
// Distance matmul via V_WMMA_F32_16X16X4_F32 (wave32 WMMA, fp32 so the
// argmin matches the fp32 reference). E (256 KB) stays resident in the
// 192 MB L2; we repack it once into WMMA B-fragment order so the hot loop
// does one coalesced b128 load per two K-steps.
// Workspace (floats): [0]=loss accum, [64..1088)=|e|^2, [2048..67584)=Bpack.
// ---------------------------------------------------------------------------

typedef float v2f __attribute__((ext_vector_type(2)));
typedef float v8f __attribute__((ext_vector_type(8)));

#define NROWS   65536
#define DDIM    64
#define KEMB    1024
#define QOUT    262144
#define WS_ESQ  64
#define WS_BPK  2048

// flat row n -> base float offset of (b, d=0, h, w) inside x / out
__device__ __forceinline__ int row_base(int n) {
    return ((n >> 10) << 16) | (n & 1023);     // b*65536 + h*32 + w
}

// --- Kernel 1: repack E into B-fragment order, precompute |e|^2, zero loss --
// Bpack layout: for tile t (16 cols), pair s2 (two K-steps), lane L holds
// float4 { (k=8*s2+2*half+0, j), (k+1, j), (k=8*s2+4+2*half, j), (k+1, j) }
// at float offset WS_BPK + ((t*8+s2)*32 + L)*4.  half = L>>4, j = 16t + L%16.
// This matches the ISA 16x4 fp32 A layout mirrored onto B (VGPR0=K0|K2,
// VGPR1=K1|K3) so a single b128 feeds two wmma K-steps per lane.
__global__ __launch_bounds__(256) void vq_prepare(const float* __restrict__ E,
                                                  float* wsf) {
    int o = blockIdx.x * 256 + threadIdx.x;          // 0 .. 65535
    {
        int q  = o & 3;                 // float within the b128
        int L  = (o >> 2) & 31;         // lane
        int st = o >> 7;                // t*8 + s2
        int s2 = st & 7, t = st >> 3;
        int s  = 2 * s2 + (q >> 1);     // K-step 0..15
        int k  = 4 * s + 2 * (L >> 4) + (q & 1);
        int j  = 16 * t + (L & 15);
        wsf[WS_BPK + o] = E[k * KEMB + j];
    }
    if (o < KEMB) {
        float acc = 0.f;
        for (int d = 0; d < DDIM; ++d) { float e = E[d * KEMB + o]; acc += e * e; }
        wsf[WS_ESQ + o] = acc;
    }
    if (o == 0) wsf[0] = 0.f;           // loss accumulator (fresh every launch)
}

// --- Kernel 2: distances via WMMA, argmin, gather, losses ------------------
__global__ __launch_bounds__(256) void vq_main(const float* __restrict__ x,
                                               const float* __restrict__ E,
                                               float* wsf,
                                               float* __restrict__ out) {
    __shared__ int lds_idx[128];
    const int lane = threadIdx.x & 31;
    const int wave = threadIdx.x >> 5;
    const int half = lane >> 4;
    const int m    = lane & 15;
    const int rowWG = blockIdx.x * 128;

    // ================= phase 1: per-wave 16 rows x 1024 codes ==============
    {
        const int rowBase = rowWG + wave * 16;
        const int n  = rowBase + m;                 // A-row handled by lane pair
        const int xb = row_base(n);

        // A fragments (ISA 16x4 fp32 layout): a[s][0]=k(4s+2*half), a[s][1]=k+1
        v2f a[16];
        float xsqp = 0.f;
#pragma unroll
        for (int s = 0; s < 16; ++s) {
            int k0 = 4 * s + 2 * half;
            float v0 = x[xb + k0 * KEMB];
            float v1 = x[xb + (k0 + 1) * KEMB];
            a[s][0] = v0; a[s][1] = v1;
            xsqp += v0 * v0 + v1 * v1;
        }
        // full |x_row|^2: lane L and L^16 together hold all 64 dims of row m
        float xsq = xsqp + __shfl_xor(xsqp, 16, 32);
        // broadcast into C-layout: VGPR v of C holds row (v + 8*half)
        float sq[8];
#pragma unroll
        for (int i = 0; i < 8; ++i) sq[i] = __shfl(xsq, half * 8 + i, 32);

        float minv[8]; int mini[8];
#pragma unroll
        for (int v = 0; v < 8; ++v) { minv[v] = 3.402823466e+38f; mini[v] = 0; }

        const float4* bp   = (const float4*)(wsf + WS_BPK);
        const float*  esqp = wsf + WS_ESQ;

        for (int t = 0; t < 64; ++t) {              // 64 tiles of 16 codes
            float esq = esqp[t * 16 + m];
            v8f c = {0.f, 0.f, 0.f, 0.f, 0.f, 0.f, 0.f, 0.f};
#pragma unroll
            for (int s2 = 0; s2 < 8; ++s2) {        // 16 chained wmma (K=64)
                float4 bv = bp[(t * 8 + s2) * 32 + lane];
                v2f b0; b0[0] = bv.x; b0[1] = bv.y;
                v2f b1; b1[0] = bv.z; b1[1] = bv.w;
                c = __builtin_amdgcn_wmma_f32_16x16x4_f32(
                        false, a[2 * s2],     false, b0, (short)0, c, false, false);
                c = __builtin_amdgcn_wmma_f32_16x16x4_f32(
                        false, a[2 * s2 + 1], false, b1, (short)0, c, false, false);
            }
            int jcol = t * 16 + m;
#pragma unroll
            for (int v = 0; v < 8; ++v) {           // running argmin, C layout
                float d = sq[v] - 2.f * c[v] + esq;
                bool take = d < minv[v];            // strict: keeps first index
                minv[v] = take ? d : minv[v];
                mini[v] = take ? jcol : mini[v];
            }
        }

        // cross-lane argmin within each 16-lane half; tie -> smaller index
#pragma unroll
        for (int off = 1; off <= 8; off <<= 1) {
#pragma unroll
            for (int v = 0; v < 8; ++v) {
                float pv = __shfl_xor(minv[v], off, 32);
                int   pi = __shfl_xor(mini[v], off, 32);
                bool take = (pv < minv[v]) || (pv == minv[v] && pi < mini[v]);
                minv[v] = take ? pv : minv[v];
                mini[v] = take ? pi : mini[v];
            }
        }
        // lane (half*16 + v), v<8 owns row (v + 8*half)
        int sel = mini[0];
#pragma unroll
        for (int v = 1; v < 8; ++v) sel = (m == v) ? mini[v] : sel;
        if (m < 8) lds_idx[wave * 16 + half * 8 + m] = sel;
    }
    __syncthreads();

    // ====== phase 2: gather codewords, straight-through output, losses =====
    {
        const int r  = threadIdx.x & 127;           // row within WG
        const int dg = threadIdx.x >> 7;            // dim half 0/1
        const int n  = rowWG + r;
        const int idx = lds_idx[r];
        const int xb  = row_base(n);
        float acc = 0.f;
#pragma unroll
        for (int i = 0; i < 32; ++i) {
            int d    = dg * 32 + i;
            float q  = E[d * KEMB + idx];
            float xv = x[xb + d * KEMB];
            float diff = q - xv;
            out[xb + d * KEMB] = xv + diff;         // x + (q - x), as reference
            acc += diff * diff;                     // (x - q)^2
        }
        if (dg == 0) out[QOUT + 2 + n] = (float)idx;   // indices (as f32)
        const int lane2 = threadIdx.x & 31;
#pragma unroll
        for (int off = 16; off >= 1; off >>= 1) acc += __shfl_xor(acc, off, 32);
        if (lane2 == 0) atomicAdd(wsf, acc);
    }
}

// --- Kernel 3: finalize the two (numerically identical) mean losses --------
__global__ void vq_finalize(const float* wsf, float* out) {
    float loss = wsf[0] * (1.0f / ((float)NROWS * (float)DDIM));
    out[QOUT]     = loss;   // dictionary_loss
    out[QOUT + 1] = loss;   // commitment_loss
}

extern "C" void kernel_launch(void* const* d_in, const int* in_sizes, int n_in,
                              void* d_out, int out_size, void* d_ws, size_t ws_size,
                              hipStream_t stream) {
    const float* x = (const float*)d_in[0];           // [64,64,32,32]
    const float* E = (const float*)d_in[1];           // [64,1024]
    float* out = (float*)d_out;
    float* wsf = (float*)d_ws;                        // needs ~264 KB

    vq_prepare<<<256, 256, 0, stream>>>(E, wsf);
    vq_main<<<512, 256, 0, stream>>>(x, E, wsf, out);
    vq_finalize<<<1, 1, 0, stream>>>(wsf, out);
}